// UncompressTransformLayer_56736517980425
// MI455X (gfx1250) — compile-verified
//
#include <hip/hip_runtime.h>
#include <math.h>

typedef float f32x4 __attribute__((ext_vector_type(4)));

// Expand packed strict-upper-triangle (row-major packing, k=1) into dense n x n.
// One thread produces 4 consecutive output columns of one row (16B NT store).
// Grid: x covers columns in 1024-wide tiles (256 thr * 4 cols), y = row.
__global__ __launch_bounds__(256) void triu_expand_nt_kernel(
    const float* __restrict__ comp,
    float* __restrict__ out,
    int n)
{
    const unsigned row = blockIdx.y;                       // wave-uniform -> SALU
    const unsigned c0  = (blockIdx.x * 256u + threadIdx.x) * 4u;
    if (c0 >= (unsigned)n) return;

    // Start of this row's packed data: row*(n-1) - row*(row-1)/2   (uniform)
    const unsigned rowoff = row * (unsigned)(n - 1) - ((row * (row - 1u)) >> 1);
    // comp index for column c (c > row): rowoff + c - row - 1
    const unsigned base_m1 = rowoff - row - 1u;            // uniform; add c per lane

    f32x4 v;
    if (c0 > row) {
        // Entire 4-wide vector strictly above the diagonal: 4 contiguous elems.
        const float* p = comp + (base_m1 + c0);
        __builtin_prefetch(p + 4096, 0, 0);                // global_prefetch_b8, stream ahead 16KB
        v.x = __builtin_nontemporal_load(p + 0);
        v.y = __builtin_nontemporal_load(p + 1);
        v.z = __builtin_nontemporal_load(p + 2);
        v.w = __builtin_nontemporal_load(p + 3);
    } else if (c0 + 3u <= row) {
        // Entire vector on/below the diagonal: zeros.
        v = (f32x4){0.0f, 0.0f, 0.0f, 0.0f};
    } else {
        // The single straddling vector of this row.
        #pragma unroll
        for (int i = 0; i < 4; ++i) {
            const unsigned c = c0 + (unsigned)i;
            v[i] = (c > row) ? __builtin_nontemporal_load(comp + base_m1 + c) : 0.0f;
        }
    }

    f32x4* dst = (f32x4*)(out + (size_t)row * (size_t)n + (size_t)c0);
    __builtin_nontemporal_store(v, dst);                   // global_store_b128 th:NT
}

extern "C" void kernel_launch(void* const* d_in, const int* in_sizes, int n_in,
                              void* d_out, int out_size, void* d_ws, size_t ws_size,
                              hipStream_t stream)
{
    (void)n_in; (void)d_ws; (void)ws_size;

    const float* comp = (const float*)d_in[0];
    float* out = (float*)d_out;

    // L = n(n-1)/2  ->  n = round(sqrt(2L)) + 1   (matches the reference)
    const long long L = in_sizes[0];
    const int n = (int)llround(sqrt(2.0 * (double)L)) + 1;   // 8192 for L=33550336

    const int cols_per_block = 256 * 4;                       // 1024
    dim3 block(256, 1, 1);
    dim3 grid((unsigned)((n + cols_per_block - 1) / cols_per_block), (unsigned)n, 1);

    triu_expand_nt_kernel<<<grid, block, 0, stream>>>(comp, out, n);
}